// GACFV1_48687749267744
// MI455X (gfx1250) — compile-verified
//
#include <hip/hip_runtime.h>
#include <hip/hip_bf16.h>

typedef __attribute__((ext_vector_type(2))) float v2f;
typedef __attribute__((ext_vector_type(8))) float v8f;

#define USER_NUM 30000
#define ITEM_NUM 70000
#define NNODES   100000
#define NNZ_E    300000
#define BATCH    16384

// ---------------------------------------------------------------------------
// zero-fill (graph-capture-safe replacement for hipMemsetAsync)
// ---------------------------------------------------------------------------
__global__ void fill0_kernel(float* __restrict__ p, long n) {
  long i = (long)blockIdx.x * blockDim.x + threadIdx.x;
  if (i < n) p[i] = 0.0f;
}

// ---------------------------------------------------------------------------
// SpMM: one wave per edge.  outA += val * X[col], outB += val * X[col]^2
// (fuses the F*F squaring into the gather so F^2 is never materialized)
// ---------------------------------------------------------------------------
__global__ void spmm_kernel(const int* __restrict__ erow,
                            const int* __restrict__ ecol,
                            const float* __restrict__ eval,
                            const float* __restrict__ X,
                            float* __restrict__ outA,
                            float* __restrict__ outB,
                            int D, int nnz) {
  long gid = (long)blockIdx.x * blockDim.x + threadIdx.x;
  int e = (int)(gid >> 5);
  int lane = (int)(gid & 31);
  if (e >= nnz) return;
  int r = erow[e];
  int c = ecol[e];
  float v = eval[e];
  const float* xc = X + (long)c * D;
  float* oa = outA + (long)r * D;
  float* ob = outB + (long)r * D;
  for (int d = lane; d < D; d += 32) {
    float x = xc[d];
    atomicAdd(&oa[d], v * x);
    atomicAdd(&ob[d], v * x * x);
  }
}

// ---------------------------------------------------------------------------
// WMMA fp32 GEMM:  Y[nrows, M] = act( X[nrows, K] @ W[M, K]^T + bias (+ Y) )
//   - block = 64-row tile (4 x 16-row sub-tiles, register-blocked per wave)
//   - wave w = output columns [16w, 16w+16), 4 accumulators per wave
//   - all 16 B-frags of a K-chunk preloaded to registers (batched DS loads)
//   - A-frags double-buffered across k-steps: loads for kk+4 are issued
//     before the wmmas of kk, so no ds_load->wmma back-to-back wait
// A frag (16x4 f32): lanes 0-15 hold K=kk,kk+1; lanes 16-31 hold K=kk+2,kk+3
// B frag (4x16 f32): lane half selects K pair, lane&15 selects output column
// D frag (16x16 f32): vgpr v -> row v (lanes<16) / row v+8 (lanes>=16)
// ---------------------------------------------------------------------------
__global__ void wmma_gemm_kernel(const float* __restrict__ X,
                                 const float* __restrict__ W,
                                 const float* __restrict__ bias,
                                 float* __restrict__ Y,
                                 long nrows, int K, int M,
                                 int doAcc, int doRelu) {
  __shared__ float Xs[64][65];
  __shared__ float Ws[128][65];

  const int tid  = threadIdx.x;
  const int nthr = blockDim.x;
  const int wave = tid >> 5;
  const int lane = tid & 31;
  const int l15  = lane & 15;
  const int half = lane >> 4;   // 0: K pair (0,1), 1: K pair (2,3)
  const long row0 = (long)blockIdx.x * 64;
  const int  col0 = wave * 16;

  v8f acc[4] = {v8f{}, v8f{}, v8f{}, v8f{}};

  for (int kc = 0; kc < K; kc += 64) {
    // cooperative coalesced staging of the X row-tile and W column-panel
    for (int idx = tid; idx < 64 * 64; idx += nthr) {
      int r = idx >> 6, cc = idx & 63;
      long gr = row0 + r;
      Xs[r][cc] = (gr < nrows) ? X[gr * (long)K + kc + cc] : 0.0f;
    }
    for (int idx = tid; idx < M * 64; idx += nthr) {
      int r = idx >> 6, cc = idx & 63;
      Ws[r][cc] = W[(long)r * K + kc + cc];
    }
    __syncthreads();

    // hoist the whole B panel slice for this wave's 16 columns (32 VGPRs)
    v2f bf[16];
#pragma unroll
    for (int i = 0; i < 16; ++i) {
      const int k2 = i * 4 + 2 * half;
      bf[i].x = Ws[col0 + l15][k2];
      bf[i].y = Ws[col0 + l15][k2 + 1];
    }

    // software-pipelined A-frags: issue loads for step i+1 before wmmas of i
    v2f aCur[4], aNxt[4];
#pragma unroll
    for (int t = 0; t < 4; ++t) {
      aCur[t].x = Xs[t * 16 + l15][2 * half];
      aCur[t].y = Xs[t * 16 + l15][2 * half + 1];
    }
#pragma unroll
    for (int i = 0; i < 16; ++i) {
      if (i < 15) {
        const int k2n = (i + 1) * 4 + 2 * half;
#pragma unroll
        for (int t = 0; t < 4; ++t) {
          aNxt[t].x = Xs[t * 16 + l15][k2n];
          aNxt[t].y = Xs[t * 16 + l15][k2n + 1];
        }
      }
#pragma unroll
      for (int t = 0; t < 4; ++t) {
        // 8 args: (neg_a, A, neg_b, B, c_mod, C, reuse_a, reuse_b)
        acc[t] = __builtin_amdgcn_wmma_f32_16x16x4_f32(
            false, aCur[t], false, bf[i], (short)0, acc[t], false, false);
      }
#pragma unroll
      for (int t = 0; t < 4; ++t) aCur[t] = aNxt[t];
    }
    __syncthreads();
  }

  const float bcol = bias[col0 + l15];
#pragma unroll
  for (int t = 0; t < 4; ++t) {
    for (int v = 0; v < 8; ++v) {
      long r = row0 + t * 16 + v + half * 8;
      if (r >= nrows) continue;
      long o = r * M + col0 + l15;
      float val = acc[t][v] + bcol;
      if (doAcc)  val += Y[o];
      if (doRelu) val = fmaxf(val, 0.0f);
      Y[o] = val;
    }
  }
}

// ---------------------------------------------------------------------------
// Gather kernel: build e[b, 896] = concat(final[u], final[USER_NUM+it])
// final columns: [0,256)->F0, [256,384)->F1, [384,448)->F2
// ---------------------------------------------------------------------------
__global__ void gather_e_kernel(const int* __restrict__ userIdx,
                                const int* __restrict__ itemIdx,
                                const float* __restrict__ F0,
                                const float* __restrict__ F1,
                                const float* __restrict__ F2,
                                float* __restrict__ E) {
  long idx = (long)blockIdx.x * blockDim.x + threadIdx.x;
  if (idx >= (long)BATCH * 896) return;
  int b = (int)(idx / 896);
  int d = (int)(idx % 896);
  int node;
  int dd;
  if (d < 448) { node = userIdx[b];            dd = d;       }
  else         { node = USER_NUM + itemIdx[b]; dd = d - 448; }
  float v;
  if (dd < 256)      v = F0[(long)node * 256 + dd];
  else if (dd < 384) v = F1[(long)node * 128 + (dd - 256)];
  else               v = F2[(long)node * 64  + (dd - 384)];
  E[idx] = v;
}

// ---------------------------------------------------------------------------
// Tail MLP: h2 = relu(T2 @ h1 + b2); out = T3 . h2 + b3   (tiny, VALU)
// ---------------------------------------------------------------------------
__global__ void mlp_tail_kernel(const float* __restrict__ h1,
                                const float* __restrict__ T2W,
                                const float* __restrict__ T2b,
                                const float* __restrict__ T3W,
                                const float* __restrict__ T3b,
                                float* __restrict__ out) {
  int b = blockIdx.x * blockDim.x + threadIdx.x;
  if (b >= BATCH) return;
  const float* h = h1 + (long)b * 64;
  float acc3 = T3b[0];
  for (int j = 0; j < 32; ++j) {
    const float* w = T2W + j * 64;
    float a = T2b[j];
    for (int k = 0; k < 64; ++k) a += w[k] * h[k];
    a = fmaxf(a, 0.0f);
    acc3 += T3W[j] * a;
  }
  out[b] = acc3;
}

// ---------------------------------------------------------------------------
extern "C" void kernel_launch(void* const* d_in, const int* in_sizes, int n_in,
                              void* d_out, int out_size, void* d_ws, size_t ws_size,
                              hipStream_t stream) {
  const int*   userIdx = (const int*)d_in[0];
  const int*   itemIdx = (const int*)d_in[1];
  const int*   erow    = (const int*)d_in[2];
  const int*   ecol    = (const int*)d_in[3];
  const float* eval    = (const float*)d_in[4];
  const float* uEmbd   = (const float*)d_in[5];
  const float* iEmbd   = (const float*)d_in[6];
  const float* W1_0 = (const float*)d_in[7];
  const float* b1_0 = (const float*)d_in[8];
  const float* W2_0 = (const float*)d_in[9];
  const float* b2_0 = (const float*)d_in[10];
  const float* W1_1 = (const float*)d_in[11];
  const float* b1_1 = (const float*)d_in[12];
  const float* W2_1 = (const float*)d_in[13];
  const float* b2_1 = (const float*)d_in[14];
  const float* T1_W = (const float*)d_in[15];
  const float* T1_b = (const float*)d_in[16];
  const float* T2_W = (const float*)d_in[17];
  const float* T2_b = (const float*)d_in[18];
  const float* T3_W = (const float*)d_in[19];
  const float* T3_b = (const float*)d_in[20];
  float* out = (float*)d_out;

  // workspace layout (floats)
  float* ws = (float*)d_ws;
  const long N = NNODES;
  float* F0 = ws;                   // N*256
  float* SA = F0 + N * 256;         // N*256  (later reused as E [B,896])
  float* SB = SA + N * 256;         // N*256  (later reused as H1 [B,64])
  float* F1 = SB + N * 256;         // N*128
  float* F2 = F1 + N * 128;         // N*64
  float* E  = SA;                    // B*896 <= N*256
  float* H1 = SB;                    // B*64

  const int rowBlocksN = (int)((N + 63) / 64);       // 1563
  const int rowBlocksB = BATCH / 64;                 // 256

  // 1. F0 = concat(uEmbd, iEmbd)
  hipMemcpyAsync(F0, uEmbd, (size_t)USER_NUM * 256 * sizeof(float),
                 hipMemcpyDeviceToDevice, stream);
  hipMemcpyAsync(F0 + (long)USER_NUM * 256, iEmbd,
                 (size_t)ITEM_NUM * 256 * sizeof(float),
                 hipMemcpyDeviceToDevice, stream);

  // 2. SA = F0 (identity term of L+I); SB = 0
  hipMemcpyAsync(SA, F0, (size_t)N * 256 * sizeof(float),
                 hipMemcpyDeviceToDevice, stream);
  {
    long n = N * 256;
    fill0_kernel<<<(int)((n + 255) / 256), 256, 0, stream>>>(SB, n);
  }

  // 3. SA += spmm(F0), SB += spmm(F0^2)
  {
    long thr = (long)NNZ_E * 32;
    spmm_kernel<<<(int)((thr + 255) / 256), 256, 0, stream>>>(
        erow, ecol, eval, F0, SA, SB, 256, NNZ_E);
  }

  // 4-5. F1 = SA @ W1_0^T + b1_0;  F1 += SB @ W2_0^T + b2_0
  wmma_gemm_kernel<<<rowBlocksN, 256, 0, stream>>>(SA, W1_0, b1_0, F1, N, 256, 128, 0, 0);
  wmma_gemm_kernel<<<rowBlocksN, 256, 0, stream>>>(SB, W2_0, b2_0, F1, N, 256, 128, 1, 0);

  // 6. layer 1 identity + zero
  hipMemcpyAsync(SA, F1, (size_t)N * 128 * sizeof(float),
                 hipMemcpyDeviceToDevice, stream);
  {
    long n = N * 128;
    fill0_kernel<<<(int)((n + 255) / 256), 256, 0, stream>>>(SB, n);
  }

  // 7. SA += spmm(F1), SB += spmm(F1^2)
  {
    long thr = (long)NNZ_E * 32;
    spmm_kernel<<<(int)((thr + 255) / 256), 256, 0, stream>>>(
        erow, ecol, eval, F1, SA, SB, 128, NNZ_E);
  }

  // 8-9. F2 = SA @ W1_1^T + b1_1;  F2 += SB @ W2_1^T + b2_1
  wmma_gemm_kernel<<<rowBlocksN, 128, 0, stream>>>(SA, W1_1, b1_1, F2, N, 128, 64, 0, 0);
  wmma_gemm_kernel<<<rowBlocksN, 128, 0, stream>>>(SB, W2_1, b2_1, F2, N, 128, 64, 1, 0);

  // 10. gather e[B,896] from F0/F1/F2 (virtual `final`, never materialized)
  {
    long n = (long)BATCH * 896;
    gather_e_kernel<<<(int)((n + 255) / 256), 256, 0, stream>>>(
        userIdx, itemIdx, F0, F1, F2, E);
  }

  // 11. H1 = relu(E @ T1_W^T + T1_b)   [B,896]x[896,64]
  wmma_gemm_kernel<<<rowBlocksB, 128, 0, stream>>>(E, T1_W, T1_b, H1, BATCH, 896, 64, 0, 1);

  // 12. out = (relu(H1 @ T2^T + b2)) @ T3^T + b3
  mlp_tail_kernel<<<(BATCH + 255) / 256, 256, 0, stream>>>(
      H1, T2_W, T2_b, T3_W, T3_b, out);
}